// ChannelWiseConv_6305011990798
// MI455X (gfx1250) — compile-verified
//
#include <hip/hip_runtime.h>
#include <math.h>

#define IN_CH 13
#define HIDDEN 20
#define NCLS 10
#define NB 8          // NBASIS = GRID + K = 8
#define KDIM1 120     // 13 + 13*8 = 117 -> padded to 120 (mult of 4)
#define KDIM2 180     // 20 + 20*8 = 180 (already mult of 4)

typedef __attribute__((ext_vector_type(2))) float v2f;
typedef __attribute__((ext_vector_type(8))) float v8f;

#if defined(__gfx1250__) && __has_builtin(__builtin_amdgcn_global_load_async_to_lds_b128) && __has_builtin(__builtin_amdgcn_s_wait_asynccnt)
#define USE_ASYNC_LDS 1
typedef int b128_t __attribute__((vector_size(16)));
typedef __attribute__((address_space(1))) b128_t* gptr_b128;
typedef __attribute__((address_space(3))) b128_t* lptr_b128;
#else
#define USE_ASYNC_LDS 0
#endif

// ---------------------------------------------------------------------------
// Kernel 1: fused 6-level depthwise conv pyramid. One block per (sample, ch).
// 64x64 tile -> LDS (async) -> 32 -> 16 -> 8 -> 4 -> 2 -> 1 scalar.
// ---------------------------------------------------------------------------
__global__ void __launch_bounds__(128) conv_pyramid_kernel(
    const float* __restrict__ x,
    const float* __restrict__ w1, const float* __restrict__ b1,
    const float* __restrict__ w2, const float* __restrict__ b2,
    const float* __restrict__ w3, const float* __restrict__ b3,
    const float* __restrict__ w4, const float* __restrict__ b4,
    const float* __restrict__ w5, const float* __restrict__ b5,
    const float* __restrict__ w6, const float* __restrict__ b6,
    float* __restrict__ h1out)
{
    __shared__ __align__(16) float sA[64 * 64];
    __shared__ __align__(16) float sB[32 * 32];

    const int bid    = blockIdx.x;
    const int sample = bid / IN_CH;
    const int ch     = bid - sample * IN_CH;
    const int tid    = threadIdx.x;

    const float* src = x + (size_t)(sample * IN_CH + ch) * 4096;

    // ---- stage 64x64 tile into LDS ----
#if USE_ASYNC_LDS
    {
        #pragma unroll
        for (int j = 0; j < 8; ++j) {
            const int idx = tid + 128 * j;   // float4 index, coalesced
            __builtin_amdgcn_global_load_async_to_lds_b128(
                (gptr_b128)(src + 4 * idx),
                (lptr_b128)(sA + 4 * idx),
                0, 0);
        }
        __builtin_amdgcn_s_wait_asynccnt(0);
    }
#else
    {
        const float4* s4 = (const float4*)src;
        float4*       d4 = (float4*)sA;
        #pragma unroll
        for (int j = 0; j < 8; ++j) {
            const int idx = tid + 128 * j;
            d4[idx] = s4[idx];
        }
    }
#endif
    __syncthreads();

    const float* wp[5] = {w1, w2, w3, w4, w5};
    const float* bp[5] = {b1, b2, b3, b4, b5};

    int W = 64;
    #pragma unroll
    for (int l = 0; l < 5; ++l) {
        const float* rd = (l & 1) ? sB : sA;
        float*       wr = (l & 1) ? sA : sB;
        float wk[9];
        #pragma unroll
        for (int k = 0; k < 9; ++k) wk[k] = wp[l][ch * 9 + k];
        const float bv = bp[l][ch];
        const int OW = W >> 1;
        for (int idx = tid; idx < OW * OW; idx += 128) {
            const int oy = idx / OW;
            const int ox = idx - oy * OW;
            float acc = bv;
            #pragma unroll
            for (int ky = 0; ky < 3; ++ky) {
                const int iy = 2 * oy - 1 + ky;
                if (iy < 0 || iy >= W) continue;
                #pragma unroll
                for (int kx = 0; kx < 3; ++kx) {
                    const int ix = 2 * ox - 1 + kx;
                    if (ix < 0 || ix >= W) continue;
                    acc = fmaf(rd[iy * W + ix], wk[ky * 3 + kx], acc);
                }
            }
            wr[idx] = fmaxf(acc, 0.0f);   // ReLU after convs 1..5
        }
        __syncthreads();
        W = OW;
    }
    // final 2x2 conv, stride 1, no pad, no relu; 2x2 data lives in sB
    if (tid == 0) {
        float r = b6[ch];
        r = fmaf(sB[0], w6[ch * 4 + 0], r);
        r = fmaf(sB[1], w6[ch * 4 + 1], r);
        r = fmaf(sB[2], w6[ch * 4 + 2], r);
        r = fmaf(sB[3], w6[ch * 4 + 3], r);
        h1out[sample * IN_CH + ch] = r;
    }
}

// ---------------------------------------------------------------------------
// KAN helpers
// ---------------------------------------------------------------------------
__device__ __forceinline__ float silu_f(float v) {
    return v / (1.0f + __expf(-v));
}

// Uniform grid: g[j] = (j-3)*0.4 - 1.0, j = 0..11. Cubic B-spline, 8 bases.
__device__ __forceinline__ void bspl8(float x, float* o) {
    float b[11];
    #pragma unroll
    for (int j = 0; j < 11; ++j) {
        const float g0 = (float)(j - 3) * 0.4f - 1.0f;
        const float g1 = (float)(j - 2) * 0.4f - 1.0f;
        b[j] = (x >= g0 && x < g1) ? 1.0f : 0.0f;
    }
    #pragma unroll
    for (int p = 1; p <= 3; ++p) {
        const float inv = 1.0f / (0.4f * (float)p);
        #pragma unroll
        for (int j = 0; j <= 10 - p; ++j) {
            const float gj   = (float)(j - 3) * 0.4f - 1.0f;       // g[j]
            const float gjp1 = (float)(j + p - 2) * 0.4f - 1.0f;   // g[j+p+1]
            b[j] = (x - gj) * inv * b[j] + (gjp1 - x) * inv * b[j + 1];
        }
    }
    #pragma unroll
    for (int n = 0; n < 8; ++n) o[n] = b[n];
}

// ---------------------------------------------------------------------------
// Kernel 2: both KAN layers + log_softmax. One wave per 16 samples.
// GEMM via V_WMMA_F32_16X16X4_F32 with features [silu(x) | bsplines(x)].
// ---------------------------------------------------------------------------
__global__ void __launch_bounds__(32) kan_head_kernel(
    const float* __restrict__ h1,
    const float* __restrict__ coef1, const float* __restrict__ sb1,
    const float* __restrict__ ss1,   const float* __restrict__ bias1,
    const float* __restrict__ coef2, const float* __restrict__ sb2,
    const float* __restrict__ ss2,   const float* __restrict__ bias2,
    float* __restrict__ out)
{
    __shared__ float sW1[KDIM1 * 32];     // folded layer-1 weights [K][32]
    __shared__ float sF1[16 * KDIM1];     // layer-1 features      [16][K]
    __shared__ float sW2[KDIM2 * 16];     // folded layer-2 weights [K][16]
    __shared__ float sF2[16 * KDIM2];     // layer-2 features      [16][K]
    __shared__ float sO1[16 * HIDDEN];    // hidden activations
    __shared__ float sLg[16 * 16];        // logits

    const int lane = threadIdx.x;
    const int s0   = blockIdx.x * 16;

    // ---- build folded W1: rows 0..12 = sb1; rows 13..116 = coef1*ss1 ----
    for (int e = lane; e < KDIM1 * 32; e += 32) {
        const int k = e >> 5, o = e & 31;
        float v = 0.0f;
        if (o < HIDDEN) {
            if (k < IN_CH) {
                v = sb1[k * HIDDEN + o];
            } else if (k < IN_CH + IN_CH * NB) {
                const int r = k - IN_CH, i = r >> 3, n = r & 7;
                v = coef1[(i * HIDDEN + o) * NB + n] * ss1[i * HIDDEN + o];
            }
        }
        sW1[e] = v;
    }
    // ---- build folded W2 ----
    for (int e = lane; e < KDIM2 * 16; e += 32) {
        const int k = e >> 4, o = e & 15;
        float v = 0.0f;
        if (o < NCLS) {
            if (k < HIDDEN) {
                v = sb2[k * NCLS + o];
            } else {
                const int r = k - HIDDEN, i = r >> 3, n = r & 7;
                v = coef2[(i * NCLS + o) * NB + n] * ss2[i * NCLS + o];
            }
        }
        sW2[e] = v;
    }
    // ---- layer-1 features ----
    for (int e = lane; e < 16 * KDIM1; e += 32) sF1[e] = 0.0f;
    __syncthreads();
    for (int t = lane; t < 16 * IN_CH; t += 32) {
        const int s = t / IN_CH, i = t - s * IN_CH;
        const float xv = h1[(s0 + s) * IN_CH + i];
        sF1[s * KDIM1 + i] = silu_f(xv);
        float bs[8]; bspl8(xv, bs);
        #pragma unroll
        for (int n = 0; n < 8; ++n) sF1[s * KDIM1 + IN_CH + i * NB + n] = bs[n];
    }
    __syncthreads();

    const int hv  = lane >> 4;   // half of wave -> K-pair / M-row split
    const int l16 = lane & 15;

    // ---- GEMM1: [16 x 120] @ [120 x 32] (cols 20..31 are zero pad) ----
    v8f clo, chi;
    {
        const float blo = bias1[l16];
        const float bhi = (16 + l16 < HIDDEN) ? bias1[16 + l16] : 0.0f;
        #pragma unroll
        for (int r = 0; r < 8; ++r) { clo[r] = blo; chi[r] = bhi; }
    }
    for (int k = 0; k < KDIM1; k += 4) {
        const int kk = k + 2 * hv;
        v2f a;  a.x  = sF1[l16 * KDIM1 + kk];      a.y  = sF1[l16 * KDIM1 + kk + 1];
        v2f bL; bL.x = sW1[kk * 32 + l16];         bL.y = sW1[(kk + 1) * 32 + l16];
        v2f bH; bH.x = sW1[kk * 32 + 16 + l16];    bH.y = sW1[(kk + 1) * 32 + 16 + l16];
        clo = __builtin_amdgcn_wmma_f32_16x16x4_f32(false, a, false, bL, (short)0, clo, false, false);
        chi = __builtin_amdgcn_wmma_f32_16x16x4_f32(false, a, false, bH, (short)0, chi, false, false);
    }
    #pragma unroll
    for (int r = 0; r < 8; ++r) {
        const int row = r + 8 * hv;
        sO1[row * HIDDEN + l16] = clo[r];
        if (16 + l16 < HIDDEN) sO1[row * HIDDEN + 16 + l16] = chi[r];
    }
    __syncthreads();

    // ---- layer-2 features ----
    for (int t = lane; t < 16 * HIDDEN; t += 32) {
        const int s = t / HIDDEN, i = t - s * HIDDEN;
        const float xv = sO1[s * HIDDEN + i];
        sF2[s * KDIM2 + i] = silu_f(xv);
        float bs[8]; bspl8(xv, bs);
        #pragma unroll
        for (int n = 0; n < 8; ++n) sF2[s * KDIM2 + HIDDEN + i * NB + n] = bs[n];
    }
    __syncthreads();

    // ---- GEMM2: [16 x 180] @ [180 x 16] (cols 10..15 zero) ----
    v8f c2;
    {
        const float bv = (l16 < NCLS) ? bias2[l16] : 0.0f;
        #pragma unroll
        for (int r = 0; r < 8; ++r) c2[r] = bv;
    }
    for (int k = 0; k < KDIM2; k += 4) {
        const int kk = k + 2 * hv;
        v2f a; a.x = sF2[l16 * KDIM2 + kk]; a.y = sF2[l16 * KDIM2 + kk + 1];
        v2f b; b.x = sW2[kk * 16 + l16];    b.y = sW2[(kk + 1) * 16 + l16];
        c2 = __builtin_amdgcn_wmma_f32_16x16x4_f32(false, a, false, b, (short)0, c2, false, false);
    }
    #pragma unroll
    for (int r = 0; r < 8; ++r) sLg[(r + 8 * hv) * 16 + l16] = c2[r];
    __syncthreads();

    // ---- log_softmax over 10 classes; lanes 0..15 own one sample each ----
    if (lane < 16) {
        float v[NCLS];
        float m = -1e30f;
        #pragma unroll
        for (int o = 0; o < NCLS; ++o) { v[o] = sLg[lane * 16 + o]; m = fmaxf(m, v[o]); }
        float ssum = 0.0f;
        #pragma unroll
        for (int o = 0; o < NCLS; ++o) ssum += __expf(v[o] - m);
        const float lse = m + __logf(ssum);
        #pragma unroll
        for (int o = 0; o < NCLS; ++o) out[(s0 + lane) * NCLS + o] = v[o] - lse;
    }
}

// ---------------------------------------------------------------------------
extern "C" void kernel_launch(void* const* d_in, const int* in_sizes, int n_in,
                              void* d_out, int out_size, void* d_ws, size_t ws_size,
                              hipStream_t stream) {
    const float* x  = (const float*)d_in[0];
    const float* w1 = (const float*)d_in[1];  const float* b1 = (const float*)d_in[2];
    const float* w2 = (const float*)d_in[3];  const float* b2 = (const float*)d_in[4];
    const float* w3 = (const float*)d_in[5];  const float* b3 = (const float*)d_in[6];
    const float* w4 = (const float*)d_in[7];  const float* b4 = (const float*)d_in[8];
    const float* w5 = (const float*)d_in[9];  const float* b5 = (const float*)d_in[10];
    const float* w6 = (const float*)d_in[11]; const float* b6 = (const float*)d_in[12];
    // d_in[13] = grid1 (uniform, hardcoded), d_in[18] = grid2 (same)
    const float* coef1 = (const float*)d_in[14];
    const float* sb1   = (const float*)d_in[15];
    const float* ss1   = (const float*)d_in[16];
    const float* bias1 = (const float*)d_in[17];
    const float* coef2 = (const float*)d_in[19];
    const float* sb2   = (const float*)d_in[20];
    const float* ss2   = (const float*)d_in[21];
    const float* bias2 = (const float*)d_in[22];

    const int batch = in_sizes[0] / (IN_CH * 64 * 64);   // 2048
    float* h1 = (float*)d_ws;                            // [batch][13]

    conv_pyramid_kernel<<<batch * IN_CH, 128, 0, stream>>>(
        x, w1, b1, w2, b2, w3, b3, w4, b4, w5, b5, w6, b6, h1);

    kan_head_kernel<<<batch / 16, 32, 0, stream>>>(
        h1, coef1, sb1, ss1, bias1, coef2, sb2, ss2, bias2, (float*)d_out);
}